// BeamSearchPostProcessModule_22634477650194
// MI455X (gfx1250) — compile-verified
//
#include <hip/hip_runtime.h>
#include <hip/hip_bf16.h>
#include <math.h>

// ---------------------------------------------------------------------------
// BeamSearchPostProcess for MI455X (gfx1250).
// HBM-optimal: logits (103 MB) are read from HBM exactly once, async-staged
// into the 320KB-per-WGP LDS (CDNA5-only capacity). The logsumexp and top-16
// scans are FUSED into a single LDS sweep that is PIPELINED under the async
// DMA via an ASYNCcnt wait ladder (each wave's async loads retire in order,
// and each thread reads only chunks it issued itself).
// ---------------------------------------------------------------------------

#define EOS_TOKEN_ID 2
#define PAD_TOKEN_ID 0
#define NUM_BEAMS    8
#define BATCH_N      64
#define VOCAB        50257
#define ROWS         (BATCH_N * NUM_BEAMS)   // 512
#define K2           (2 * NUM_BEAMS)         // 16 candidates per row / per batch
#define TPB_A        256
#define V4           (VOCAB / 4)             // 12564 float4 chunks (1 tail float)
#define NCH          ((V4 + TPB_A - 1) / TPB_A)  // 50 uniform async issues/thread

typedef int v4i_b128 __attribute__((vector_size(4 * sizeof(int))));
typedef __attribute__((address_space(1))) v4i_b128 gas_v4i;   // global
typedef __attribute__((address_space(3))) v4i_b128 las_v4i;   // LDS

#if __has_builtin(__builtin_amdgcn_global_load_async_to_lds_b128)
#define HAVE_ASYNC_LDS 1
#else
#define HAVE_ASYNC_LDS 0
#endif

#if __has_builtin(__builtin_amdgcn_s_wait_asynccnt)
#define WAIT_ASYNC(n) __builtin_amdgcn_s_wait_asynccnt(n)
#else
#define WAIT_ASYNC(n) asm volatile("s_wait_asynccnt %0" ::"i"(n) : "memory")
#endif

// ---------------------------------------------------------------------------
// Kernel A: one workgroup per (batch,beam) row.
//   1) issue exactly NCH async b128 loads per thread (uniform per wave)
//   2) single fused sweep (online logsumexp + per-thread top-16), staged
//      behind the DMA with s_wait_asynccnt 40/30/20/10/0
//   3) LDS tree-reduce of (max,sum); LDS selection of the row top-16
// Outputs per row: 16 (score,token) pairs (scores already include bias).
// ---------------------------------------------------------------------------
__global__ __launch_bounds__(TPB_A) void row_lse_topk_kernel(
    const float* __restrict__ logits,
    const float* __restrict__ beam_scores,
    float* __restrict__ row_scores,   // [ROWS][K2]
    int*   __restrict__ row_tokens)   // [ROWS][K2]
{
  __shared__ __align__(16) float lds_row[VOCAB + 3];   // 201,040 B
  __shared__ __align__(16) float dummy_lds[TPB_A * 4]; // ghost-issue target
  __shared__ float red_m[TPB_A];
  __shared__ float red_s[TPB_A];
  __shared__ int   red_i[TPB_A];
  __shared__ float s_bias;

  const int row = blockIdx.x;
  const int tid = threadIdx.x;
  const float* rp = logits + (size_t)row * VOCAB;

  // ---- Online-lse + top-16 thread state ----
  float m = -INFINITY, s = 0.f;
  float ls[K2]; int lt[K2];
#pragma unroll
  for (int i = 0; i < K2; i++) { ls[i] = -INFINITY; lt[i] = 0; }
  float lmin = -INFINITY; int lminpos = 0;

  auto consider = [&](float x, int tok) {
    if (x > lmin) {
      ls[lminpos] = x; lt[lminpos] = tok;
      float nm = ls[0]; int np = 0;
#pragma unroll
      for (int i = 1; i < K2; i++) { if (ls[i] < nm) { nm = ls[i]; np = i; } }
      lmin = nm; lminpos = np;
    }
  };

  auto sweep_chunk = [&](float4 x, int i) {
    float xm = fmaxf(fmaxf(x.x, x.y), fmaxf(x.z, x.w));
    float nm = fmaxf(m, xm);
    s = s * __expf(m - nm) +
        (__expf(x.x - nm) + __expf(x.y - nm) + __expf(x.z - nm) + __expf(x.w - nm));
    m = nm;
    int t0 = 4 * i;
    consider(x.x, t0); consider(x.y, t0 + 1);
    consider(x.z, t0 + 2); consider(x.w, t0 + 3);
  };

#if HAVE_ASYNC_LDS
  // ---- Issue exactly NCH async loads per thread (wave-uniform count) ----
  for (int j = 0; j < NCH; ++j) {
    int i = tid + j * TPB_A;
    const float* gsrc = (i < V4) ? (rp + 4 * i) : rp;            // safe ghost src
    float*       ldst = (i < V4) ? &lds_row[4 * i] : &dummy_lds[4 * tid];
    __builtin_amdgcn_global_load_async_to_lds_b128(
        (gas_v4i*)gsrc, (las_v4i*)ldst, 0, 0);
  }

  // ---- Fused sweep, pipelined behind the DMA (ASYNCcnt ladder) ----
  // After s_wait_asynccnt <= NCH-n, this wave's first n issues are landed;
  // each thread reads only chunks it issued itself, so this is race-free.
#define PROC_RANGE(J0, J1)                                   \
  for (int j = (J0); j < (J1); ++j) {                        \
    int i = tid + j * TPB_A;                                 \
    if (i < V4) sweep_chunk(((const float4*)lds_row)[i], i); \
  }
  WAIT_ASYNC(40); PROC_RANGE(0, 10)
  WAIT_ASYNC(30); PROC_RANGE(10, 20)
  WAIT_ASYNC(20); PROC_RANGE(20, 30)
  WAIT_ASYNC(10); PROC_RANGE(30, 40)
  WAIT_ASYNC(0);  PROC_RANGE(40, NCH)
#undef PROC_RANGE
#else
  // Fallback: fused sweep straight from global (no staging).
  for (int i = tid; i < V4; i += TPB_A)
    sweep_chunk(((const float4*)rp)[i], i);
#endif

  // ---- Tail element (VOCAB-1), read directly from global by its owner ----
  if (tid == (V4 % TPB_A)) {
    float x = rp[VOCAB - 1];
    float nm = fmaxf(m, x);
    s = s * __expf(m - nm) + __expf(x - nm);
    m = nm;
    consider(x, VOCAB - 1);
  }

  // ---- Block reduce of (max, sum) -> bias ----
  red_m[tid] = m; red_s[tid] = s;
  __syncthreads();
  for (int off = TPB_A / 2; off > 0; off >>= 1) {
    if (tid < off) {
      float m2 = red_m[tid + off], s2 = red_s[tid + off];
      float M = fmaxf(red_m[tid], m2);
      red_s[tid] = red_s[tid] * __expf(red_m[tid] - M) + s2 * __expf(m2 - M);
      red_m[tid] = M;
    }
    __syncthreads();
  }
  if (tid == 0) s_bias = beam_scores[row] - (red_m[0] + __logf(red_s[0]));
  __syncthreads();
  const float bias = s_bias;

  // ---- Reduce 256 x 16 candidates -> row top-16 (sorted desc, low-idx ties)
  // (safe to overwrite lds_row: all sweeps completed before the barriers above)
  float* cs = lds_row;                       // 4096 scores (aliases row buffer)
  int*   ct = (int*)(lds_row + TPB_A * K2);  // 4096 tokens
#pragma unroll
  for (int i = 0; i < K2; i++) {
    cs[tid * K2 + i] = ls[i];
    ct[tid * K2 + i] = lt[i];
  }
  __syncthreads();

  for (int r = 0; r < K2; r++) {
    float bm = -INFINITY; int bj = -1;
    for (int i = tid; i < TPB_A * K2; i += TPB_A) {
      float v = cs[i];
      if (v > bm || (v == bm && bj >= 0 && ct[i] < ct[bj])) { bm = v; bj = i; }
    }
    red_m[tid] = bm; red_i[tid] = bj;
    __syncthreads();
    for (int off = TPB_A / 2; off > 0; off >>= 1) {
      if (tid < off) {
        float v2 = red_m[tid + off]; int j2 = red_i[tid + off];
        float v1 = red_m[tid];       int j1 = red_i[tid];
        bool take = (v2 > v1) ||
                    (v2 == v1 && j2 >= 0 && (j1 < 0 || ct[j2] < ct[j1]));
        if (take) { red_m[tid] = v2; red_i[tid] = j2; }
      }
      __syncthreads();
    }
    if (tid == 0) {
      int w = (red_i[0] < 0) ? 0 : red_i[0];
      row_scores[row * K2 + r] = red_m[0] + bias;
      row_tokens[row * K2 + r] = ct[w];
      cs[w] = -INFINITY;
    }
    __syncthreads();
  }
}

// ---------------------------------------------------------------------------
// Kernel B: one block per batch. Merge the 8 rows' top-16 -> batch top-16
// (the batch top-16 is contained in the union), then exact process() +
// degenerate hypothesis update (<=8 EOS pushes all have length<K, so the
// scoreboard reduces to worst=min(worst,score/denom), length++).
// Output layout (all fp32): [0,512) nb_scores | [512,1024) nb_tokens |
// [1024,1536) nb_indices | [1536,1600) done_new.
// ---------------------------------------------------------------------------
__global__ __launch_bounds__(128) void batch_merge_kernel(
    const float* __restrict__ row_scores,
    const int*   __restrict__ row_tokens,
    const unsigned char* __restrict__ done_in,
    const int*   __restrict__ cur_len_p,
    float* __restrict__ out)
{
  __shared__ float bs_[128];
  __shared__ int   bi_[128];
  __shared__ float h_ts[K2];
  __shared__ int   h_ti[K2];

  const int batch = blockIdx.x;
  const int tid = threadIdx.x;

  {
    int beam = tid / K2;                      // 0..7
    int e    = tid % K2;                      // 0..15
    int r    = batch * NUM_BEAMS + beam;
    bs_[tid] = row_scores[r * K2 + e];
    bi_[tid] = beam * VOCAB + row_tokens[r * K2 + e];  // flat idx in [0, K*V)
  }
  __syncthreads();

  if (tid == 0) {
    // top-16 of 128, sorted descending, ties -> lower flat index (lax.top_k)
    for (int r = 0; r < K2; r++) {
      float bm = bs_[0]; int bj = 0;
      for (int j = 1; j < 128; j++) {
        float v = bs_[j];
        if (v > bm || (v == bm && bi_[j] < bi_[bj])) { bm = v; bj = j; }
      }
      h_ts[r] = bm; h_ti[r] = bi_[bj];
      bs_[bj] = -INFINITY;
    }

    const bool  dn        = (done_in[batch] != 0);
    const float denom     = powf((float)cur_len_p[0], 1.0f);  // LENGTH_PENALTY=1
    const float cur_score = h_ts[0] / denom;                  // max, sorted desc

    float nbs[NUM_BEAMS]; int nbt[NUM_BEAMS]; int nbi[NUM_BEAMS];
    for (int c = 0; c < NUM_BEAMS; c++) { nbs[c] = 0.f; nbt[c] = 0; nbi[c] = 0; }

    int ne = 0, length = 0;
    float worst = 1e9f;
    for (int rank = 0; rank < K2; rank++) {
      float sc  = h_ts[rank];
      int   fi  = h_ti[rank];
      int   beam = fi / VOCAB;
      int   tok  = fi - beam * VOCAB;
      bool  eos    = (tok == EOS_TOKEN_ID);
      bool  active = (ne < NUM_BEAMS);
      if (active && !eos) {
        nbs[ne] = sc; nbt[ne] = tok; nbi[ne] = batch * NUM_BEAMS + beam;
      }
      if (active && eos && rank < NUM_BEAMS && !dn) {
        float sh = sc / denom;
        if (length < NUM_BEAMS || sh > worst) {
          worst = fminf(worst, sh);
          length++;
        }
      }
      if (!eos) ne++;
    }
    bool done_new = dn || ((length >= NUM_BEAMS) && (worst >= cur_score));

    for (int c = 0; c < NUM_BEAMS; c++) {
      int o = batch * NUM_BEAMS + c;
      out[o]            = dn ? 0.f : nbs[c];
      out[ROWS + o]     = dn ? (float)PAD_TOKEN_ID : (float)nbt[c];
      out[2 * ROWS + o] = dn ? 0.f : (float)nbi[c];
    }
    out[3 * ROWS + batch] = done_new ? 1.f : 0.f;
  }
}

// ---------------------------------------------------------------------------
// Inputs (setup_inputs order):
//   d_in[0] input_ids [512*256] i32 (UNUSED: does not affect any output)
//   d_in[1] next_token_logits [512*50257] f32
//   d_in[2] beam_scores [512] f32
//   d_in[3] done [64] bool (byte)
//   d_in[4] cur_len scalar i32
// d_out: 1600 f32 (nb_scores | nb_tokens | nb_indices | done_new, flattened)
// ---------------------------------------------------------------------------
extern "C" void kernel_launch(void* const* d_in, const int* in_sizes, int n_in,
                              void* d_out, int out_size, void* d_ws, size_t ws_size,
                              hipStream_t stream) {
  (void)in_sizes; (void)n_in; (void)out_size; (void)ws_size;

  const float*         logits      = (const float*)d_in[1];
  const float*         beam_scores = (const float*)d_in[2];
  const unsigned char* done        = (const unsigned char*)d_in[3];
  const int*           cur_len     = (const int*)d_in[4];

  float* row_scores = (float*)d_ws;                                      // 32 KB
  int*   row_tokens = (int*)((char*)d_ws + (size_t)ROWS * K2 * sizeof(float));

  row_lse_topk_kernel<<<ROWS, TPB_A, 0, stream>>>(logits, beam_scores,
                                                  row_scores, row_tokens);
  batch_merge_kernel<<<BATCH_N, 128, 0, stream>>>(row_scores, row_tokens,
                                                  done, cur_len, (float*)d_out);
}